// DummyFairGCN_38113539785179
// MI455X (gfx1250) — compile-verified
//
#include <hip/hip_runtime.h>

// ---------------------------------------------------------------------------
// GCN forward (3x GCNConv+BN+ReLU, then Linear+ReLU, Linear) for gfx1250.
// Dense GEMMs: V_WMMA_F32_16X16X4_F32, one wave computes a 16 x (16*NT) tile
// of C so the A fragment is reused NT times per k-step.
// Sparse aggregation: hardware fp32 atomics into L2-resident buffers.
// ---------------------------------------------------------------------------

#define NN 50000
#define NE 800000
#define BN_EPS 1e-5f

typedef __attribute__((ext_vector_type(2))) float v2f;
typedef __attribute__((ext_vector_type(8))) float v8f;

// ---------------------------------------------------------------------------
// WMMA GEMM: C[M x Fo] = A[M x K] @ B[K x Fo] (+ epilogue)
// f32 fragment layouts per CDNA5 ISA 7.12.2:
//   A 16x4 : lane L holds row M=L&15, elements K = 2*(L>>4) + {0,1}
//   B 4x16 : lane L holds col N=L&15, elements K = 2*(L>>4) + {0,1}
//   C 16x16: lane L holds col N=L&15, VGPR r holds row M = r + 8*(L>>4)
// OOB columns (Fo=40 case) use clamped loads (finite garbage) and masked
// stores; no zeroing needed since D's column n depends only on B's column n.
// MODE 0: C = acc   MODE 1: C = relu(acc + bias)   MODE 2: C = acc + bias
// ---------------------------------------------------------------------------
template <int K, int Fo, int MODE>
__global__ __launch_bounds__(32) void gemm_wmma(
    const float* __restrict__ A, const float* __restrict__ B,
    const float* __restrict__ bias, float* __restrict__ C)
{
    constexpr int NT = (Fo >= 64) ? 4 : ((Fo + 15) / 16);   // 16-col tiles/wave
    const int m0   = blockIdx.x << 4;
    const int n0   = blockIdx.y * (16 * NT);
    const int lane = threadIdx.x;       // 0..31
    const int half = lane >> 4;         // 0 or 1
    const int mr   = lane & 15;
    const int nc   = lane & 15;

    int ncol[NT];                       // clamped (always in-bounds) column
    #pragma unroll
    for (int t = 0; t < NT; ++t) {
        int n = n0 + 16 * t + nc;
        ncol[t] = (n < Fo) ? n : (Fo - 1);
    }

    v8f acc[NT];
    const v8f zero = {0.f, 0.f, 0.f, 0.f, 0.f, 0.f, 0.f, 0.f};
    #pragma unroll
    for (int t = 0; t < NT; ++t) acc[t] = zero;

    const float* __restrict__ arow = A + (size_t)(m0 + mr) * K;

    #pragma unroll 8
    for (int k0 = 0; k0 < K; k0 += 4) {
        const int ka = k0 + 2 * half;
        v2f a;
        a[0] = arow[ka];
        a[1] = arow[ka + 1];
        #pragma unroll
        for (int t = 0; t < NT; ++t) {
            v2f b;
            b[0] = B[(size_t)ka * Fo + ncol[t]];
            b[1] = B[(size_t)(ka + 1) * Fo + ncol[t]];
            acc[t] = __builtin_amdgcn_wmma_f32_16x16x4_f32(
                /*neg_a=*/false, a, /*neg_b=*/false, b,
                /*c_mod=*/(short)0, acc[t], /*reuse_a=*/false, /*reuse_b=*/false);
        }
    }

    #pragma unroll
    for (int t = 0; t < NT; ++t) {
        const int n = n0 + 16 * t + nc;
        if (n < Fo) {
            float bv = 0.f;
            if (MODE != 0) bv = bias[n];
            #pragma unroll
            for (int r = 0; r < 8; ++r) {
                int m = m0 + r + 8 * half;
                float v = acc[t][r];
                if (MODE == 1) { v += bv; v = v > 0.f ? v : 0.f; }
                else if (MODE == 2) { v += bv; }
                C[(size_t)m * Fo + n] = v;
            }
        }
    }
}

// ---------------------------------------------------------------------------
// Degree / normalization helpers
// ---------------------------------------------------------------------------
__global__ void k_fill(float* __restrict__ p, int n, float val)
{
    int i = blockIdx.x * blockDim.x + threadIdx.x;
    if (i < n) p[i] = val;
}

__global__ void k_degcount(const int* __restrict__ dst, float* __restrict__ deg, int e)
{
    int i = blockIdx.x * blockDim.x + threadIdx.x;
    if (i < e) unsafeAtomicAdd(&deg[dst[i]], 1.0f);
}

__global__ void k_rsqrt(float* __restrict__ p, int n)
{
    int i = blockIdx.x * blockDim.x + threadIdx.x;
    if (i < n) p[i] = rsqrtf(fmaxf(p[i], 1.0f));
}

// out[i,f] = xw[i,f] * dinv[i]^2   (self-loop contribution, also inits out)
template <int Fo>
__global__ void k_selfloop(const float* __restrict__ xw, const float* __restrict__ dinv,
                           float* __restrict__ out)
{
    int idx = blockIdx.x * blockDim.x + threadIdx.x;
    if (idx < NN * Fo) {
        int i = idx / Fo;               // Fo is a power of two -> shift
        float d = dinv[i];
        out[idx] = xw[idx] * d * d;
    }
}

// one 32-lane wave per edge: out[dst] += xw[src] * dinv[src]*dinv[dst]
template <int Fo>
__global__ void k_scatter(const float* __restrict__ xw, const int* __restrict__ src,
                          const int* __restrict__ dst, const float* __restrict__ dinv,
                          float* __restrict__ out)
{
    int warp = (blockIdx.x * blockDim.x + threadIdx.x) >> 5;
    int lane = threadIdx.x & 31;
    if (warp >= NE) return;
    int s = src[warp];
    int d = dst[warp];
    float w = dinv[s] * dinv[d];
    const float4* __restrict__ xs = (const float4*)(xw + (size_t)s * Fo);
    float* __restrict__ od = out + (size_t)d * Fo;
    #pragma unroll
    for (int it = 0; it < Fo / 128; ++it) {
        int f4 = lane + 32 * it;
        float4 v = xs[f4];
        unsafeAtomicAdd(od + 4 * f4 + 0, v.x * w);
        unsafeAtomicAdd(od + 4 * f4 + 1, v.y * w);
        unsafeAtomicAdd(od + 4 * f4 + 2, v.z * w);
        unsafeAtomicAdd(od + 4 * f4 + 3, v.w * w);
    }
}

// h = relu( ((h + conv_b) - m) * rsqrt(v+eps) * g + b )   in place
template <int Fo>
__global__ void k_bnrelu(float* __restrict__ h, const float* __restrict__ cb,
                         const float* __restrict__ g, const float* __restrict__ b,
                         const float* __restrict__ m, const float* __restrict__ v)
{
    int idx = blockIdx.x * blockDim.x + threadIdx.x;
    if (idx < NN * Fo) {
        int f = idx & (Fo - 1);         // Fo is a power of two
        float x = h[idx] + cb[f];
        x = (x - m[f]) * rsqrtf(v[f] + BN_EPS) * g[f] + b[f];
        h[idx] = fmaxf(x, 0.f);
    }
}

// ---------------------------------------------------------------------------
extern "C" void kernel_launch(void* const* d_in, const int* in_sizes, int n_in,
                              void* d_out, int out_size, void* d_ws, size_t ws_size,
                              hipStream_t stream)
{
    (void)in_sizes; (void)n_in; (void)out_size; (void)ws_size;

    const float* x   = (const float*)d_in[0];
    const int*   ei  = (const int*)d_in[1];
    const float* cw[3] = {(const float*)d_in[2],  (const float*)d_in[8],  (const float*)d_in[14]};
    const float* cb[3] = {(const float*)d_in[3],  (const float*)d_in[9],  (const float*)d_in[15]};
    const float* bg[3] = {(const float*)d_in[4],  (const float*)d_in[10], (const float*)d_in[16]};
    const float* bb[3] = {(const float*)d_in[5],  (const float*)d_in[11], (const float*)d_in[17]};
    const float* bm[3] = {(const float*)d_in[6],  (const float*)d_in[12], (const float*)d_in[18]};
    const float* bv[3] = {(const float*)d_in[7],  (const float*)d_in[13], (const float*)d_in[19]};
    const float* lw1 = (const float*)d_in[20];
    const float* lb1 = (const float*)d_in[21];
    const float* lw2 = (const float*)d_in[22];
    const float* lb2 = (const float*)d_in[23];
    float* out = (float*)d_out;

    const int* srcI = ei;        // edge_index[0]
    const int* dstI = ei + NE;   // edge_index[1]

    // workspace carve (256B aligned): dinv[N], bufA[N*256], bufB[N*256] ~103MB
    char* p = (char*)d_ws;
    auto carve = [&](size_t bytes) -> void* {
        void* r = (void*)p;
        p += (bytes + 255) & ~(size_t)255;
        return r;
    };
    float* dinv = (float*)carve((size_t)NN * 4);
    float* bufA = (float*)carve((size_t)NN * 256 * 4);
    float* bufB = (float*)carve((size_t)NN * 256 * 4);

    const int MB = NN / 16;             // 3125 row tiles

    // --- D^-1/2 for (A + I) ---
    k_fill<<<(NN + 255) / 256, 256, 0, stream>>>(dinv, NN, 1.0f);   // self-loop
    k_degcount<<<(NE + 255) / 256, 256, 0, stream>>>(dstI, dinv, NE);
    k_rsqrt<<<(NN + 255) / 256, 256, 0, stream>>>(dinv, NN);

    // --- conv1: 128 -> 128 ---
    gemm_wmma<128, 128, 0><<<dim3(MB, 2), 32, 0, stream>>>(x, cw[0], nullptr, bufA);
    k_selfloop<128><<<(NN * 128 + 255) / 256, 256, 0, stream>>>(bufA, dinv, bufB);
    k_scatter<128><<<(NE + 7) / 8, 256, 0, stream>>>(bufA, srcI, dstI, dinv, bufB);
    k_bnrelu<128><<<(NN * 128 + 255) / 256, 256, 0, stream>>>(bufB, cb[0], bg[0], bb[0], bm[0], bv[0]);

    // --- conv2: 128 -> 128 ---
    gemm_wmma<128, 128, 0><<<dim3(MB, 2), 32, 0, stream>>>(bufB, cw[1], nullptr, bufA);
    k_selfloop<128><<<(NN * 128 + 255) / 256, 256, 0, stream>>>(bufA, dinv, bufB);
    k_scatter<128><<<(NE + 7) / 8, 256, 0, stream>>>(bufA, srcI, dstI, dinv, bufB);
    k_bnrelu<128><<<(NN * 128 + 255) / 256, 256, 0, stream>>>(bufB, cb[1], bg[1], bb[1], bm[1], bv[1]);

    // --- conv3: 128 -> 256 ---
    gemm_wmma<128, 256, 0><<<dim3(MB, 4), 32, 0, stream>>>(bufB, cw[2], nullptr, bufA);
    k_selfloop<256><<<(NN * 256 + 255) / 256, 256, 0, stream>>>(bufA, dinv, bufB);
    k_scatter<256><<<(NE + 7) / 8, 256, 0, stream>>>(bufA, srcI, dstI, dinv, bufB);
    k_bnrelu<256><<<(NN * 256 + 255) / 256, 256, 0, stream>>>(bufB, cb[2], bg[2], bb[2], bm[2], bv[2]);

    // --- lin1: relu(h @ W1 + b1) -> bufA [N x 128] ---
    gemm_wmma<256, 128, 1><<<dim3(MB, 2), 32, 0, stream>>>(bufB, lw1, lb1, bufA);

    // --- lin2: bufA @ W2 + b2 -> out [N x 40] ---
    gemm_wmma<128, 40, 2><<<dim3(MB, 1), 32, 0, stream>>>(bufA, lw2, lb2, out);
}